// DecorelationNormalization_38920993636472
// MI455X (gfx1250) — compile-verified
//
#include <hip/hip_runtime.h>

// Problem constants (match reference setup_inputs)
#define Bn 32
#define Wd 64
#define Hd 64
#define Cc 256
#define Nn (Bn * Wd * Hd)   // 131072 samples per channel
#define Gg 16               // groups
#define Mm 16               // channels per group
#define EPS_ 0.001f

#define NBLK 256            // sample-slice blocks
#define SPB (Nn / NBLK)     // 512 samples per block

typedef __attribute__((ext_vector_type(2))) float v2f;
typedef __attribute__((ext_vector_type(4))) float v4f;
typedef __attribute__((ext_vector_type(8))) float v8f;

// Hardware fp32 atomic add (GLOBAL_ATOMIC_ADD_F32), relaxed / device scope.
__device__ __forceinline__ void atomic_add_f32(float* p, float v) {
    __hip_atomic_fetch_add(p, v, __ATOMIC_RELAXED, __HIP_MEMORY_SCOPE_AGENT);
}

// Workspace layout (floats):
//   [0,    256)  : per-channel sums
//   [256,  4352) : m2   = sum of x x^T per group  (G x 16 x 16)
//   [4352, 8448) : Wmat = L^{-1} per group        (G x 16 x 16)
//   [8448, 8704) : means (256)
#define WS_SUMS  0
#define WS_M2    256
#define WS_W     4352
#define WS_MEANS 8448
#define WS_ZERO_COUNT 4352   // sums + m2 must start at zero

__global__ void zero_ws(float* __restrict__ ws, int n) {
    int i = blockIdx.x * blockDim.x + threadIdx.x;
    if (i < n) ws[i] = 0.0f;
}

// Pass 1: each block = 16 waves; wave w accumulates group w's second-moment
// matrix over a 512-sample slice using V_WMMA_F32_16X16X4_F32, plus the
// per-channel sum. Partials reduced into ws with fp32 global atomics.
// Regular (RT) loads deliberately populate L2: x (128 MiB) fits in the
// 192 MB global L2 and is re-read by the whiten pass.
__global__ __launch_bounds__(512) void stats_kernel(const float* __restrict__ x,
                                                    float* __restrict__ ws) {
    float* sums = ws + WS_SUMS;
    float* m2   = ws + WS_M2;

    const int lane = threadIdx.x & 31;
    const int g    = threadIdx.x >> 5;     // wave id == group id
    const int r    = lane & 15;            // matrix row / column index
    const int half = lane >> 4;            // K-pair half of the wave (K 0,1 vs 2,3)
    const int chb  = g * Mm;               // first channel of this group
    const long n0  = (long)blockIdx.x * SPB;

    v8f acc = {};                          // 16x16 fp32 accumulator tile
    float sacc = 0.0f;

    // A-matrix 16x4 f32 layout: lanes 0-15 hold K=0 (K=1 in v1), lanes 16-31 hold K=2,3.
    // For the symmetric product F * F^T the B operand register image is identical.
    const float* xb = x + chb + r;         // this lane's channel, varying sample
    #pragma unroll 2
    for (int t = 0; t < SPB; t += 4) {
        long s = n0 + t + 2 * half;        // sample index for this lane's K pair
        v2f a;
        a.x = xb[s * Cc];
        a.y = xb[(s + 1) * Cc];
        sacc += a.x + a.y;
        acc = __builtin_amdgcn_wmma_f32_16x16x4_f32(
            /*neg_a=*/false, a, /*neg_b=*/false, a,
            /*c_mod=*/(short)0, acc, /*reuse_a=*/false, /*reuse_b=*/false);
    }

    atomic_add_f32(&sums[chb + r], sacc);

    // C/D layout: VGPR i -> (M=i, N=lane) for lanes 0-15, (M=i+8, N=lane-16) for 16-31.
    float* m2g = m2 + g * 256;
    #pragma unroll
    for (int i = 0; i < 8; ++i) {
        int m = i + 8 * half;
        atomic_add_f32(&m2g[m * 16 + r], acc[i]);
    }
}

// Pass 2: finalize covariance (faithful double /(N-1) as in the reference),
// Cholesky-factor, and invert L. 16 independent 16x16 problems -> 1 thread each.
__global__ __launch_bounds__(32) void solve_kernel(float* __restrict__ ws) {
    __shared__ float Ash[Gg][256];
    __shared__ float Wsh[Gg][256];

    const float* sums = ws + WS_SUMS;
    const float* m2   = ws + WS_M2;
    float* Wout  = ws + WS_W;
    float* means = ws + WS_MEANS;

    const int g = threadIdx.x;
    if (g < Gg) {
        const float Nf    = (float)Nn;
        const float denom = Nf - 1.0f;

        float mu[16];
        for (int m = 0; m < 16; ++m) {
            mu[m] = sums[g * 16 + m] / Nf;
            means[g * 16 + m] = mu[m];
        }

        float* A = Ash[g];
        const float* m2g = m2 + g * 256;
        for (int m = 0; m < 16; ++m)
            for (int k = 0; k < 16; ++k) {
                float cv = (m2g[m * 16 + k] - Nf * mu[m] * mu[k]) / denom;
                cv = (1.0f - EPS_) * cv + (m == k ? EPS_ : 0.0f);
                A[m * 16 + k] = cv / denom;   // reference divides by (N-1) twice
            }

        // In-place Cholesky (lower triangular).
        for (int j = 0; j < 16; ++j) {
            float d = A[j * 16 + j];
            for (int k = 0; k < j; ++k) d -= A[j * 16 + k] * A[j * 16 + k];
            float Ljj = sqrtf(d);
            A[j * 16 + j] = Ljj;
            float inv = 1.0f / Ljj;
            for (int i = j + 1; i < 16; ++i) {
                float s = A[i * 16 + j];
                for (int k = 0; k < j; ++k) s -= A[i * 16 + k] * A[j * 16 + k];
                A[i * 16 + j] = s * inv;
            }
        }

        // W = L^{-1} by forward substitution (lower triangular, zero above).
        float* Wl = Wsh[g];
        for (int j = 0; j < 16; ++j) {
            for (int i = 0; i < j; ++i) Wl[i * 16 + j] = 0.0f;
            Wl[j * 16 + j] = 1.0f / A[j * 16 + j];
            for (int i = j + 1; i < 16; ++i) {
                float s = 0.0f;
                for (int k = j; k < i; ++k) s += A[i * 16 + k] * Wl[k * 16 + j];
                Wl[i * 16 + j] = -s / A[i * 16 + i];
            }
        }
        for (int t = 0; t < 256; ++t) Wout[g * 256 + t] = Wl[t];
    }
}

// Pass 3: whiten. Each wave owns a group; 16 samples x 16 channels output tile
// per iteration via 4 chained V_WMMA_F32_16X16X4_F32 (K = 16 total).
// Output stores are NON-TEMPORAL: x is L2-resident after pass 1 (128 MiB in a
// 192 MB L2); NT stores keep the 128 MiB of output from evicting it, so the
// second read of x stays in L2 and HBM traffic is ~read-once + write-once.
__global__ __launch_bounds__(512) void whiten_kernel(const float* __restrict__ x,
                                                     const float* __restrict__ ws,
                                                     float* __restrict__ out) {
    const float* Wmat  = ws + WS_W;
    const float* means = ws + WS_MEANS;

    const int lane = threadIdx.x & 31;
    const int g    = threadIdx.x >> 5;
    const int r    = lane & 15;
    const int half = lane >> 4;
    const int chb  = g * Mm;
    const long n0  = (long)blockIdx.x * SPB;

    // Hoist the A operand (W columns) and the mean pairs out of the sample loop.
    v2f aw[4];
    v2f mu[4];
    const float* Wg = Wmat + g * 256;
    #pragma unroll
    for (int j = 0; j < 4; ++j) {
        int kb = 4 * j + 2 * half;         // K index handled by this lane half
        aw[j].x = Wg[r * 16 + kb];
        aw[j].y = Wg[r * 16 + kb + 1];
        mu[j].x = means[chb + kb];
        mu[j].y = means[chb + kb + 1];
    }

    for (int t = 0; t < SPB; t += 16) {
        long sn = n0 + t + r;              // sample column owned by this lane
        const float* xp = x + sn * Cc + chb;
        v8f c = {};
        #pragma unroll
        for (int j = 0; j < 4; ++j) {
            int kb = 4 * j + 2 * half;
            v2f b = *reinterpret_cast<const v2f*>(xp + kb);  // 8B aligned
            b.x -= mu[j].x;
            b.y -= mu[j].y;
            c = __builtin_amdgcn_wmma_f32_16x16x4_f32(
                false, aw[j], false, b, (short)0, c, false, false);
        }
        // D tile: lane half 0 writes channels [chb, chb+8), half 1 writes [chb+8, chb+16)
        float* op = out + sn * Cc + chb + 8 * half;
        v4f o0 = {c[0], c[1], c[2], c[3]};
        v4f o1 = {c[4], c[5], c[6], c[7]};
        __builtin_nontemporal_store(o0, reinterpret_cast<v4f*>(op));
        __builtin_nontemporal_store(o1, reinterpret_cast<v4f*>(op + 4));
    }
}

extern "C" void kernel_launch(void* const* d_in, const int* in_sizes, int n_in,
                              void* d_out, int out_size, void* d_ws, size_t ws_size,
                              hipStream_t stream) {
    const float* x = (const float*)d_in[0];
    float* out = (float*)d_out;
    float* ws  = (float*)d_ws;

    zero_ws<<<(WS_ZERO_COUNT + 255) / 256, 256, 0, stream>>>(ws, WS_ZERO_COUNT);
    stats_kernel<<<NBLK, 512, 0, stream>>>(x, ws);
    solve_kernel<<<1, 32, 0, stream>>>(ws);
    whiten_kernel<<<NBLK, 512, 0, stream>>>(x, ws, out);
}